// RecurrentMomentumAttention_13735305412831
// MI455X (gfx1250) — compile-verified
//
#include <hip/hip_runtime.h>

// ext-vector types (avoid HIP's struct float4 so __builtin_nontemporal_* accepts them)
typedef float f4  __attribute__((ext_vector_type(4)));
typedef float v2f __attribute__((ext_vector_type(2)));
typedef float v8f __attribute__((ext_vector_type(8)));

#define MU_F   0.9f
#define EPS_F  1e-6f

constexpr int DDIM = 64;                 // feature dim
constexpr int MDIM = 64;                 // value dim
constexpr int TILE = DDIM * MDIM;        // 4096 floats per (n,h)
constexpr int TPB  = 128;                // 4 wave32s per block
constexpr int SSTR = 72;                 // padded LDS row stride (floats): 2*72 % 64 = 16 -> no half-wave bank overlap

// elu(x)+1  ==  x+1 (x>0)  |  exp(x) (x<=0)
__device__ __forceinline__ float featmap(float x) {
    return x > 0.0f ? x + 1.0f : __expf(x);
}

__global__ __launch_bounds__(TPB)
void rma_kernel(const float* __restrict__ query,
                const float* __restrict__ key,
                const float* __restrict__ value,
                const float* __restrict__ Si,
                const float* __restrict__ Zi,
                const float* __restrict__ Pi,
                float* __restrict__ outV,
                float* __restrict__ outS,
                float* __restrict__ outZ,
                float* __restrict__ outP)
{
    __shared__ float qf[DDIM];
    __shared__ float kf[DDIM];
    __shared__ float vv[MDIM];
    __shared__ float red[DDIM];
    __shared__ float Zs;                               // 1/(Q.Zi_new + eps)
    __shared__ alignas(16) float sn_lds[DDIM * SSTR];  // Si_new tile, padded rows

    const int p = blockIdx.x;            // flat (n,h) pair
    const int t = threadIdx.x;
    const long vbase = (long)p * DDIM;   // vector base
    const long mbase = (long)p * TILE;   // matrix base

    // ---------------- stage 1: feature maps, Zi update, normalizer ----------------
    if (t < DDIM) {
        const float fq = featmap(query[vbase + t]);
        const float fk = featmap(key  [vbase + t]);
        qf[t] = fq;
        kf[t] = fk;
        vv[t] = value[vbase + t];
        const float zn = Zi[vbase + t] + fk;
        outZ[vbase + t] = zn;            // Zi_new
        red[t] = fq * zn;
    }
    __syncthreads();
    if (t == 0) {
        float s = 0.0f;
        #pragma unroll
        for (int i = 0; i < DDIM; ++i) s += red[i];
        Zs = 1.0f / (s + EPS_F);
    }

    // ---------------- stage 2: bandwidth-bound momentum stream ----------------
    // Pi_new = MU*Pi - K[d]*v[m];  Si_new = Si - Pi_new.
    // Fully coalesced b128, non-temporal (1 GB stream, zero reuse).
    #pragma unroll
    for (int c = 0; c < TILE / (TPB * 4); ++c) {       // 8 iterations
        const int  idx = c * TPB + t;                  // float4 index within tile
        const int  d   = idx >> 4;                     // row (feature dim)
        const int  m4  = (idx & 15) * 4;               // first column of this float4
        const long e   = mbase + (long)idx * 4;

        const f4 pi = __builtin_nontemporal_load((const f4*)(Pi + e));
        const f4 si = __builtin_nontemporal_load((const f4*)(Si + e));
        const float kd = kf[d];

        f4 pn, sn;
        pn.x = MU_F * pi.x - kd * vv[m4 + 0];
        pn.y = MU_F * pi.y - kd * vv[m4 + 1];
        pn.z = MU_F * pi.z - kd * vv[m4 + 2];
        pn.w = MU_F * pi.w - kd * vv[m4 + 3];
        sn.x = si.x - pn.x;
        sn.y = si.y - pn.y;
        sn.z = si.z - pn.z;
        sn.w = si.w - pn.w;

        __builtin_nontemporal_store(pn, (f4*)(outP + e));
        __builtin_nontemporal_store(sn, (f4*)(outS + e));

        // stage Si_new for the WMMA read-out (16B aligned: SSTR*4=288 and m4*4 are multiples of 16)
        *(f4*)(&sn_lds[d * SSTR + m4]) = sn;
    }
    __syncthreads();

    // ---------------- stage 3: read-out  V = Z * (Q . Si_new)  via WMMA ----------------
    // Wave w computes V[16w .. 16w+16).  A(16x4) = Q-chunk replicated on every row
    // (so all 16 rows of D are the identical V chunk); B(4x16) = Si_new slab from LDS.
    // f32 A layout: VGPR0 holds K = 2*half, VGPR1 holds K = 2*half+1 (half = lane>=16);
    // B/C/D are row-striped across lanes (col = lane%16).
    {
        const int w    = t >> 5;          // wave id 0..3
        const int l    = t & 31;          // lane
        const int half = l >> 4;          // 0: lanes 0-15, 1: lanes 16-31
        const int li   = l & 15;
        const int col  = w * 16 + li;     // output column of V this lane carries

        v8f acc = {};
        #pragma unroll
        for (int kc = 0; kc < 16; ++kc) { // K = 64 in steps of 4
            const int r0 = 4 * kc + 2 * half;
            v2f a, b;
            a.x = qf[r0];
            a.y = qf[r0 + 1];
            b.x = sn_lds[(r0    ) * SSTR + col];
            b.y = sn_lds[(r0 + 1) * SSTR + col];
            // D = A*B + C   (neg_a, A, neg_b, B, c_mod, C, reuse_a, reuse_b)
            acc = __builtin_amdgcn_wmma_f32_16x16x4_f32(false, a, false, b,
                                                        (short)0, acc, false, false);
        }
        // every row of D is identical; row 0 lives in acc[0] of lanes 0-15
        if (l < 16) {
            outV[vbase + col] = acc[0] * Zs;
        }
    }
}

extern "C" void kernel_launch(void* const* d_in, const int* in_sizes, int n_in,
                              void* d_out, int out_size, void* d_ws, size_t ws_size,
                              hipStream_t stream) {
    const float* query = (const float*)d_in[0];
    const float* key   = (const float*)d_in[1];
    const float* value = (const float*)d_in[2];
    const float* Si    = (const float*)d_in[3];
    const float* Zi    = (const float*)d_in[4];
    const float* Pi    = (const float*)d_in[5];

    const int pairs = in_sizes[0] / DDIM;          // N*H (= 16384 for the reference shapes)

    float* out = (float*)d_out;                    // V | Si_new | Zi_new | Pi_new
    float* outV = out;
    float* outS = outV + (size_t)pairs * MDIM;
    float* outZ = outS + (size_t)pairs * TILE;
    float* outP = outZ + (size_t)pairs * DDIM;

    rma_kernel<<<pairs, TPB, 0, stream>>>(query, key, value, Si, Zi, Pi,
                                          outV, outS, outZ, outP);
}